// Conv_13872744366727
// MI455X (gfx1250) — compile-verified
//
#include <hip/hip_runtime.h>

typedef __attribute__((ext_vector_type(2))) float v2f;
typedef __attribute__((ext_vector_type(8))) float v8f;

#define B_    4
#define CIN   64
#define NPTS  16384
#define NQv   4096
#define KNN   32
#define KW    16
#define HID   16
#define COUT  128
#define QT    16              // queries per workgroup tile
#define AGG_I (CIN * KW)      // 1024
#define AGG_S (AGG_I + 4)     // padded LDS stride (bank-conflict free for WMMA B reads)
#define SA_S  (KW + 1)        // padded A stride

// -------------------------------------------------------------------------
// Fused: gather -> MLP(A) -> feature/A contraction (LDS) -> WMMA f32 GEMM
// Writes pre-BatchNorm output (with conv bias) to out[B][COUT][NQ].
// -------------------------------------------------------------------------
__global__ __launch_bounds__(256)
void convpoint_fused_kernel(const float* __restrict__ input,    // [B,CIN,N]
                            const float* __restrict__ points,   // [B,3,N]
                            const float* __restrict__ support,  // [B,3,NQ]
                            const float* __restrict__ w1,       // [HID,3]
                            const float* __restrict__ b1,       // [HID]
                            const float* __restrict__ w2,       // [KW,HID]
                            const float* __restrict__ b2,       // [KW]
                            const float* __restrict__ Wm,       // [COUT, CIN*KW]
                            const float* __restrict__ bW,       // [COUT]
                            const int*   __restrict__ indices,  // [B,NQ,KNN]
                            float*       __restrict__ out)      // [B,COUT,NQ]
{
    __shared__ float sA[QT][KNN][SA_S];    // per-neighbor kernel weights A
    __shared__ float sAgg[QT][AGG_S];      // per-query 1024-vector (GEMM B operand)
    __shared__ int   sIdx[QT][KNN];
    __shared__ float sW1[HID * 3], sB1[HID], sW2[KW * HID], sB2[KW];

    const int tid = threadIdx.x;
    const int b   = blockIdx.x / (NQv / QT);
    const int q0  = (blockIdx.x % (NQv / QT)) * QT;

    // ---- stage tiny MLP weights ----
    if (tid < HID * 3)  sW1[tid] = w1[tid];
    if (tid < HID)      sB1[tid] = b1[tid];
    if (tid < KW * HID) sW2[tid] = w2[tid];
    if (tid < KW)       sB2[tid] = b2[tid];

    // ---- stage indices for this query tile ----
    for (int i = tid; i < QT * KNN; i += 256) {
        sIdx[i / KNN][i % KNN] =
            indices[(size_t)b * NQv * KNN + (size_t)(q0 + i / KNN) * KNN + (i % KNN)];
    }
    __syncthreads();

    // ---- MLP on relative positions: A[q][k][w] ----
    for (int p = tid; p < QT * KNN; p += 256) {
        const int q = p / KNN, k = p % KNN;
        const int idx = sIdx[q][k];
        const float* pb = points  + (size_t)b * 3 * NPTS;
        const float* sb = support + (size_t)b * 3 * NQv;
        const float rx = pb[0 * NPTS + idx] - sb[0 * NQv + q0 + q];
        const float ry = pb[1 * NPTS + idx] - sb[1 * NQv + q0 + q];
        const float rz = pb[2 * NPTS + idx] - sb[2 * NQv + q0 + q];
        float h[HID];
#pragma unroll
        for (int j = 0; j < HID; ++j) {
            float v = sW1[j * 3 + 0] * rx + sW1[j * 3 + 1] * ry + sW1[j * 3 + 2] * rz + sB1[j];
            h[j] = v > 0.0f ? v : 0.0f;
        }
#pragma unroll
        for (int w = 0; w < KW; ++w) {
            float a = sB2[w];
#pragma unroll
            for (int j = 0; j < HID; ++j) a += sW2[w * HID + j] * h[j];
            sA[q][k][w] = a;
        }
    }
    __syncthreads();

    // ---- agg[q][c*16+w] = (1/K) sum_k feats[c,q,k] * A[q][k][w] ----
    const float invK = 1.0f / (float)KNN;
    for (int p = tid; p < QT * CIN; p += 256) {
        const int q = p / CIN, c = p % CIN;   // q uniform per wave -> sA broadcast reads
        float acc[KW];
#pragma unroll
        for (int w = 0; w < KW; ++w) acc[w] = 0.0f;
        const float* inb = input + (size_t)b * CIN * NPTS + (size_t)c * NPTS;
#pragma unroll 4
        for (int k = 0; k < KNN; ++k) {
            const float f = inb[sIdx[q][k]];       // L2-resident gather
#pragma unroll
            for (int w = 0; w < KW; ++w) acc[w] += f * sA[q][k][w];
        }
#pragma unroll
        for (int w = 0; w < KW; ++w) sAgg[q][c * KW + w] = acc[w] * invK;
    }
    __syncthreads();

    // ---- final GEMM: out[o,q] = sum_i Wm[o,i] * agg[q,i], via v_wmma_f32_16x16x4_f32
    // Wave 'wv' owns Cout rows [wv*16, wv*16+16). ISA lane layouts (CDNA5 ISA):
    //   A 16x4 f32 : lanes 0-15 -> M=lane,   VGPR0=K0,VGPR1=K1 ; lanes 16-31 -> K2,K3
    //   B 4x16 f32 : lanes 0-15 -> N=lane,   VGPR0=K0,VGPR1=K1 ; lanes 16-31 -> K2,K3
    //   C 16x16    : lanes 0-15 rows 0-7 in v0..v7 ; lanes 16-31 rows 8-15
    const int wv      = tid >> 5;
    const int lane    = tid & 31;
    const int half    = lane >> 4;      // 0: K0/K1 rows 0-7 ; 1: K2/K3 rows 8-15
    const int l16     = lane & 15;
    const int oBlk    = wv * 16;
    const int k0      = half * 2;
    const float* Wrow = Wm + (size_t)(oBlk + l16) * AGG_I;

    v8f acc = {};
#pragma unroll 4
    for (int i = 0; i < AGG_I; i += 4) {
        v2f af, bf;
        af.x = Wrow[i + k0];
        af.y = Wrow[i + k0 + 1];
        bf.x = sAgg[l16][i + k0];       // padded stride -> conflict-free
        bf.y = sAgg[l16][i + k0 + 1];
        acc = __builtin_amdgcn_wmma_f32_16x16x4_f32(
                  false, af, false, bf, (short)0, acc, false, false);
    }

    // ---- store D + conv bias (pre-BN) ----
    const int qn = q0 + l16;
#pragma unroll
    for (int r = 0; r < 8; ++r) {
        const int o = oBlk + half * 8 + r;
        out[(size_t)b * COUT * NQv + (size_t)o * NQv + qn] = acc[r] + bW[o];
    }
}

// -------------------------------------------------------------------------
// Per-channel BN stats over (B, NQ): scale = gamma*rsqrt(var+eps),
// shift = beta - mean*scale. One block per channel.
// -------------------------------------------------------------------------
__global__ __launch_bounds__(256)
void bn_stats_kernel(const float* __restrict__ out,
                     const float* __restrict__ gamma,
                     const float* __restrict__ beta,
                     float*       __restrict__ ss)   // [2*COUT]: scale, shift
{
    __shared__ float s_sum[256], s_sq[256];
    const int o = blockIdx.x;
    float s = 0.0f, s2 = 0.0f;
    for (int i = threadIdx.x; i < B_ * NQv; i += 256) {
        const int bb = i / NQv, n = i % NQv;
        const float x = out[(size_t)bb * COUT * NQv + (size_t)o * NQv + n];
        s += x; s2 += x * x;
    }
    s_sum[threadIdx.x] = s;
    s_sq[threadIdx.x]  = s2;
    __syncthreads();
    for (int st = 128; st > 0; st >>= 1) {
        if (threadIdx.x < st) {
            s_sum[threadIdx.x] += s_sum[threadIdx.x + st];
            s_sq[threadIdx.x]  += s_sq[threadIdx.x + st];
        }
        __syncthreads();
    }
    if (threadIdx.x == 0) {
        const float inv_n = 1.0f / (float)(B_ * NQv);
        const float mean  = s_sum[0] * inv_n;
        const float var   = s_sq[0] * inv_n - mean * mean;
        const float inv   = rsqrtf(var + 1e-5f);
        const float sc    = gamma[o] * inv;
        ss[o]        = sc;
        ss[COUT + o] = beta[o] - mean * sc;
    }
}

// -------------------------------------------------------------------------
// In-place BN apply + ReLU.
// -------------------------------------------------------------------------
__global__ __launch_bounds__(256)
void bn_relu_kernel(float* __restrict__ out, const float* __restrict__ ss)
{
    const size_t i = (size_t)blockIdx.x * 256 + threadIdx.x;
    const int o = (int)((i / NQv) % COUT);
    const float y = out[i] * ss[o] + ss[COUT + o];
    out[i] = y > 0.0f ? y : 0.0f;
}

// -------------------------------------------------------------------------
extern "C" void kernel_launch(void* const* d_in, const int* in_sizes, int n_in,
                              void* d_out, int out_size, void* d_ws, size_t ws_size,
                              hipStream_t stream)
{
    const float* input   = (const float*)d_in[0];
    const float* points  = (const float*)d_in[1];
    const float* support = (const float*)d_in[2];
    const float* w1      = (const float*)d_in[3];
    const float* b1      = (const float*)d_in[4];
    const float* w2      = (const float*)d_in[5];
    const float* b2      = (const float*)d_in[6];
    const float* Wm      = (const float*)d_in[7];
    const float* bW      = (const float*)d_in[8];
    const float* gamma   = (const float*)d_in[9];
    const float* beta    = (const float*)d_in[10];
    const int*   indices = (const int*)d_in[11];

    float* out = (float*)d_out;
    float* ss  = (float*)d_ws;    // 2*COUT floats

    convpoint_fused_kernel<<<B_ * (NQv / QT), 256, 0, stream>>>(
        input, points, support, w1, b1, w2, b2, Wm, bW, indices, out);

    bn_stats_kernel<<<COUT, 256, 0, stream>>>(out, gamma, beta, ss);

    bn_relu_kernel<<<(B_ * COUT * NQv) / 256, 256, 0, stream>>>(out, ss);

    // Tuple tail: (support_points, indices) copied raw after the conv output.
    char* tail = (char*)d_out + (size_t)B_ * COUT * NQv * sizeof(float);
    hipMemcpyAsync(tail, support, (size_t)in_sizes[2] * sizeof(float),
                   hipMemcpyDeviceToDevice, stream);
    hipMemcpyAsync(tail + (size_t)in_sizes[2] * sizeof(float), indices,
                   (size_t)in_sizes[11] * sizeof(int),
                   hipMemcpyDeviceToDevice, stream);
}